// FPN_62062277427557
// MI455X (gfx1250) — compile-verified
//
#include <hip/hip_runtime.h>

// Problem constants (from reference)
#define BATCH 2
#define NPTS  8192
#define KNN   16
#define D1    128
#define D2    256
#define DC    384   // D1 + D2
#define DOUT  128

typedef __attribute__((ext_vector_type(2))) float        v2f;
typedef __attribute__((ext_vector_type(8))) float        v8f;
typedef __attribute__((ext_vector_type(4))) unsigned int v4u;
typedef __attribute__((ext_vector_type(4))) int          v4i;
typedef __attribute__((ext_vector_type(8))) int          v8i;

// Padded-LDS geometry for TDM-staged W:
//   pad_interval = 128 dwords (code 6), pad_amount = 1 dword (code 0)
//   row = 384 dwords = 3*128  ->  effective row stride 387 dwords
//   387 mod 64 = 3 (coprime with 64 banks) -> conflict-free B reads
#define WROW_DW   387
#define WLDS_DW   (49152 + 384)            // 49536 dwords
#define WLDS_BYTES (WLDS_DW * 4)           // 198144 B (< 320KB WGP LDS)

// ---------------------------------------------------------------------------
// Kernel 1: brute-force KNN (top-16 smallest sq-dists) + inverse-distance
// weights. One thread per query; keys streamed through LDS in SoA tiles.
// Top-16 lives entirely in VGPRs (static indexing only).
// ---------------------------------------------------------------------------
__global__ __launch_bounds__(128) void fpn_knn_kernel(
    const float* __restrict__ xyz1, const float* __restrict__ xyz2,
    int* __restrict__ outIdx, float* __restrict__ outW)
{
    __shared__ float sx[256], sy[256], sz[256];

    const int blocksPerBatch = NPTS / 128;
    const int b = blockIdx.x / blocksPerBatch;
    const int q = (blockIdx.x % blocksPerBatch) * 128 + threadIdx.x;

    const float* qp = xyz2 + ((size_t)(b * NPTS + q)) * 3;
    const float qx = qp[0], qy = qp[1], qz = qp[2];

    float bd[KNN];
    int   bi[KNN];
#pragma unroll
    for (int s = 0; s < KNN; ++s) { bd[s] = 3.4e38f; bi[s] = 0; }
    float cmax = 3.4e38f;
    int   cslot = 0;

    for (int t0 = 0; t0 < NPTS; t0 += 256) {
        // cooperative SoA load of 256 key points (2 per thread)
        {
            const float* p = xyz1 + ((size_t)(b * NPTS + t0 + threadIdx.x)) * 3;
            sx[threadIdx.x] = p[0]; sy[threadIdx.x] = p[1]; sz[threadIdx.x] = p[2];
            p += 128 * 3;
            sx[threadIdx.x + 128] = p[0];
            sy[threadIdx.x + 128] = p[1];
            sz[threadIdx.x + 128] = p[2];
        }
        __syncthreads();

#pragma unroll 4
        for (int j = 0; j < 256; ++j) {
            float dx = qx - sx[j];
            float dy = qy - sy[j];
            float dz = qz - sz[j];
            float d = fmaf(dx, dx, fmaf(dy, dy, dz * dz));
            if (d < cmax) {
                // replace current max slot (static-index unrolled write)
#pragma unroll
                for (int s = 0; s < KNN; ++s)
                    if (s == cslot) { bd[s] = d; bi[s] = t0 + j; }
                // rescan for new max
                cmax = -1.0f;
#pragma unroll
                for (int s = 0; s < KNN; ++s)
                    if (bd[s] > cmax) { cmax = bd[s]; cslot = s; }
            }
        }
        __syncthreads();
    }

    // inverse-distance weights, normalized
    float rw[KNN];
    float ssum = 0.0f;
#pragma unroll
    for (int s = 0; s < KNN; ++s) { rw[s] = 1.0f / (bd[s] + 1e-8f); ssum += rw[s]; }
    const float inv = 1.0f / ssum;

    const size_t base = ((size_t)(b * NPTS + q)) * KNN;
#pragma unroll
    for (int s = 0; s < KNN; ++s) {
        outIdx[base + s] = bi[s];
        outW[base + s]   = rw[s] * inv;
    }
}

// ---------------------------------------------------------------------------
// Kernel 2: weighted gather-interpolation from points2 + concat with points1.
// One wave32 per query, lanes striped over channels -> every gathered row of
// points2 (1KB) is a fully coalesced load. Writes cat[B*N, 384].
// ---------------------------------------------------------------------------
__global__ __launch_bounds__(256) void fpn_interp_cat_kernel(
    const float* __restrict__ points1, const float* __restrict__ points2,
    const int* __restrict__ idx, const float* __restrict__ w,
    float* __restrict__ cat)
{
    const int lane = threadIdx.x & 31;
    const int q = blockIdx.x * 8 + (threadIdx.x >> 5);  // global query over B*N
    const int b = q >> 13;                              // q / NPTS

    float4 a0 = {0.f, 0.f, 0.f, 0.f};
    float4 a1 = {0.f, 0.f, 0.f, 0.f};

    const size_t wbase = (size_t)q * KNN;
#pragma unroll
    for (int k = 0; k < KNN; ++k) {
        const int   ki = idx[wbase + k];   // uniform across wave (broadcast)
        const float kw = w[wbase + k];
        const float4* row =
            (const float4*)(points2 + ((size_t)(b * NPTS + ki)) * D2);
        float4 v0 = row[lane * 2];
        float4 v1 = row[lane * 2 + 1];
        a0.x = fmaf(kw, v0.x, a0.x); a0.y = fmaf(kw, v0.y, a0.y);
        a0.z = fmaf(kw, v0.z, a0.z); a0.w = fmaf(kw, v0.w, a0.w);
        a1.x = fmaf(kw, v1.x, a1.x); a1.y = fmaf(kw, v1.y, a1.y);
        a1.z = fmaf(kw, v1.z, a1.z); a1.w = fmaf(kw, v1.w, a1.w);
    }

    float* crow = cat + (size_t)q * DC;
    ((float4*)(crow + D1))[lane * 2]     = a0;   // interp -> cols 128..383
    ((float4*)(crow + D1))[lane * 2 + 1] = a1;
    // points1 copy -> cols 0..127 (128 floats, 4 per lane)
    ((float4*)crow)[lane] = ((const float4*)(points1 + (size_t)q * D1))[lane];
}

// ---------------------------------------------------------------------------
// Kernel 3: pointwise linear as f32 WMMA GEMM, with the whole W matrix
// staged into LDS by the Tensor Data Mover (TDM).
//   out[16384,128] = cat[16384,384] x W^T[384,128] + bias
//
// TDM D# describes W as a 2D tensor (dim0=384 elems of 4B, dim1=128 rows,
// tile = whole tensor, stride0=384) with LDS padding of 1 dword per 128
// dwords stored -> padded row stride 387 dwords -> conflict-free ds_load_b64
// of the B operand pair {W[o][k], W[o][k+1]} (B layout: lane->N=o, comps->K).
// Wave 0 issues the TDM op (EXEC is ignored for tensor ops), waits on
// TENSORcnt, and __syncthreads releases the other 7 waves.
// ---------------------------------------------------------------------------
__global__ __launch_bounds__(256) void fpn_gemm_wmma_kernel(
    const float* __restrict__ cat, const float* __restrict__ W,
    const float* __restrict__ bias, float* __restrict__ out)
{
    extern __shared__ __align__(16) char smem_raw[];
    const float* sWf = (const float*)smem_raw;   // padded W, 49536 dwords

    const int tid = threadIdx.x;

    if (tid < 32) {  // wave 0 uniformly: issue one TDM load of all of W
        const unsigned long long ga = (unsigned long long)W;
        const unsigned int ldsAddr =
            (unsigned int)(unsigned long long)(uintptr_t)&smem_raw[0];

        // ---- D# group 0 (128b): count=1 | lds_addr | global_addr | type=2
        v4u g0;
        g0.x = 0x1u;                               // count=1, user descriptor
        g0.y = ldsAddr;                            // LDS byte address
        g0.z = (unsigned int)ga;                   // global_addr[31:0]
        g0.w = (unsigned int)((ga >> 32) & 0x1FFFFFFull) | (2u << 30); // [56:32]|type=2

        // ---- D# group 1 (256b)
        // w0: data_size=2 (4B) | pad_enable | pad_interval=6 (128dw) | pad_amount=0 (1dw)
        v8i g1;
        g1.s0 = (int)((2u << 16) | (1u << 20) | (6u << 22));
        g1.s1 = (int)((unsigned)(DC & 0xFFFF) << 16);        // tensor_dim0 lo16 @63:48
        g1.s2 = (int)(((unsigned)DC >> 16) | ((unsigned)(DOUT & 0xFFFF) << 16)); // dim0 hi | dim1 lo
        g1.s3 = (int)(((unsigned)DOUT >> 16) | ((unsigned)DC << 16));  // dim1 hi | tile_dim0
        g1.s4 = (int)(DOUT);                                  // tile_dim1 | tile_dim2=0
        g1.s5 = (int)(DC);                                    // tensor_dim0_stride lo32
        g1.s6 = 0;                                            // stride0 hi | stride1 lo
        g1.s7 = 0;                                            // stride1 hi
        // ---- groups 2/3: zero (2D tensor; tile_dim2=0 => unused)
        v4i gz = {0, 0, 0, 0};

#if defined(__clang_major__) && (__clang_major__ >= 23)
        v8i gz8 = {0, 0, 0, 0, 0, 0, 0, 0};
        __builtin_amdgcn_tensor_load_to_lds(g0, g1, gz, gz, gz8, 0);
#else
        __builtin_amdgcn_tensor_load_to_lds(g0, g1, gz, gz, 0);
#endif
        __builtin_amdgcn_s_wait_tensorcnt(0);
    }
    __syncthreads();

    const int lane = tid & 31;
    const int l15  = lane & 15;
    const int hi   = lane >> 4;
    const int wave = tid >> 5;
    const int m0   = blockIdx.x * 128 + wave * 16;

    // A operand: lane L -> M=L&15, comps = K {2*(L>=16), +1} (32-bit A 16x4)
    const v2f* arow = (const v2f*)(cat + (size_t)(m0 + l15) * DC);

    // Per-lane padded-LDS row bases for the 8 N-tiles (dword units)
    int baseo[8];
#pragma unroll
    for (int nt = 0; nt < 8; ++nt)
        baseo[nt] = (nt * 16 + l15) * WROW_DW;

    v8f c[8];
#pragma unroll
    for (int nt = 0; nt < 8; ++nt) {
        const float bv = bias[nt * 16 + l15];   // C layout: N = lane&15 only
        c[nt] = (v8f){bv, bv, bv, bv, bv, bv, bv, bv};
    }

    for (int k0 = 0; k0 < DC; k0 += 4) {
        const int k    = k0 + 2 * hi;
        const int koff = k + (k >> 7);          // pad-corrected dword offset
        const v2f a = arow[(k0 >> 1) + hi];
#pragma unroll
        for (int nt = 0; nt < 8; ++nt) {
            const v2f bb = *(const v2f*)(sWf + baseo[nt] + koff);
            c[nt] = __builtin_amdgcn_wmma_f32_16x16x4_f32(
                false, a, false, bb, (short)0, c[nt], false, false);
        }
    }

#pragma unroll
    for (int nt = 0; nt < 8; ++nt) {
        const int o = nt * 16 + l15;
#pragma unroll
        for (int r = 0; r < 8; ++r) {
            // C/D layout: element r of lane L -> M = r + 8*(L>=16), N = L&15
            out[(size_t)(m0 + r + 8 * hi) * DOUT + o] = c[nt][r];
        }
    }
}

// ---------------------------------------------------------------------------
// Launch: KNN -> interp/concat -> TDM-staged WMMA GEMM. Workspace layout:
//   [0, 1MB)      idx   int  [B*N,16]
//   [1MB, 2MB)    w     f32  [B*N,16]
//   [2MB, ~27MB)  cat   f32  [B*N,384]
// ---------------------------------------------------------------------------
extern "C" void kernel_launch(void* const* d_in, const int* in_sizes, int n_in,
                              void* d_out, int out_size, void* d_ws, size_t ws_size,
                              hipStream_t stream)
{
    const float* xyz1    = (const float*)d_in[0];
    const float* xyz2    = (const float*)d_in[1];
    const float* points1 = (const float*)d_in[2];
    const float* points2 = (const float*)d_in[3];
    const float* Wm      = (const float*)d_in[4];
    const float* bvec    = (const float*)d_in[5];
    float* out = (float*)d_out;

    char* ws = (char*)d_ws;
    const size_t idxBytes = (size_t)BATCH * NPTS * KNN * sizeof(int);
    int*   wsIdx = (int*)ws;
    float* wsW   = (float*)(ws + idxBytes);
    float* wsCat = (float*)(ws + 2 * idxBytes);

    fpn_knn_kernel<<<BATCH * (NPTS / 128), 128, 0, stream>>>(xyz1, xyz2, wsIdx, wsW);

    fpn_interp_cat_kernel<<<(BATCH * NPTS) / 8, 256, 0, stream>>>(
        points1, points2, wsIdx, wsW, wsCat);

    fpn_gemm_wmma_kernel<<<(BATCH * NPTS) / 128, 256, WLDS_BYTES, stream>>>(
        wsCat, Wm, bvec, out);
}